// PositionAwareGraphAttention_36275293782541
// MI455X (gfx1250) — compile-verified
//
#include <hip/hip_runtime.h>
#include <hip/hip_bf16.h>
#include <stdint.h>

typedef __attribute__((ext_vector_type(2))) float v2f;
typedef __attribute__((ext_vector_type(8))) float v8f;

#define BM 64
#define BN 128
#define BK 32

// A tile: row stride 36 floats (144B: 16B-aligned rows, conflict-free frag reads)
#define AS_STRIDE 36
// B tile: K-pair interleaved: BsP[k>>1][col*2 + (k&1)], inner stride 288 dwords
// (288 mod 64 == 32 -> the two K-halves of a wave hit disjoint bank halves)
#define BS_STRIDE 288

// Generic row-major f32 WMMA GEMM: C[M,N] = A[M,K] @ B[K,N]
// Optional epilogue: C[row,col] += pos_table[positions[row] * N + col]
// Batched via blockIdx.z with element strides aBatch/bBatch/cBatch.
__global__ __launch_bounds__(256)
void gemm_wmma_f32(const float* __restrict__ A, const float* __restrict__ B,
                   float* __restrict__ C, int M, int N, int K,
                   long long aBatch, long long bBatch, long long cBatch,
                   const int* __restrict__ positions,
                   const float* __restrict__ pos_table)
{
    __shared__ float As[BM][AS_STRIDE];    // 64 x 36 floats = 9216 B
    __shared__ float BsP[BK / 2][BS_STRIDE]; // 16 x 288 floats = 18432 B

    const int bz = blockIdx.z;
    A += (long long)bz * aBatch;
    B += (long long)bz * bBatch;
    C += (long long)bz * cBatch;

    const int tid  = threadIdx.x;
    const int lane = tid & 31;
    const int w    = tid >> 5;     // wave id 0..7
    const int wy   = w & 3;        // 4 waves along M (4*16 = 64)
    const int wx   = w >> 2;       // 2 waves along N (2*64 = 128)
    const int lo   = lane & 15;    // M (A) / N (B,C) index within tile
    const int hi   = lane >> 4;    // selects K-half (A,B) / M-half (C)

    const int m0 = blockIdx.x * BM;
    const int n0 = blockIdx.y * BN;

    v8f acc[4] = {};               // wave computes a 16x64 strip (4 tiles of 16x16)

    for (int k0 = 0; k0 < K; k0 += BK) {
        // ---- stage A tile (64x32) via CDNA5 async global->LDS DMA ----
        // 512 float4 slots, 2 per thread; per-lane LDS dest addr + 64-bit global addr.
        #pragma unroll
        for (int i = 0; i < 2; ++i) {
            int s = tid + i * 256;
            int r = s >> 3, c = (s & 7) << 2;
            const float* g = &A[(long long)(m0 + r) * K + (k0 + c)];
            unsigned lds_addr = (unsigned)(uintptr_t)&As[r][c];
            asm volatile("global_load_async_to_lds_b128 %0, %1, off"
                         :: "v"(lds_addr), "v"((unsigned long long)(uintptr_t)g)
                         : "memory");
        }
        // ---- stage B tile (32x128) into K-pair interleaved layout ----
        // 1024 float4 slots, 4 per thread
        #pragma unroll
        for (int i = 0; i < 4; ++i) {
            int s = tid + i * 256;
            int r = s >> 5, c = (s & 31) << 2;
            float4 v = *reinterpret_cast<const float4*>(
                &B[(long long)(k0 + r) * N + (n0 + c)]);
            float* dst = &BsP[r >> 1][2 * c + (r & 1)];
            dst[0] = v.x; dst[2] = v.y; dst[4] = v.z; dst[6] = v.w;
        }
        // wait for this wave's async DMA into LDS, then block-sync
        asm volatile("s_wait_asynccnt 0x0" ::: "memory");
        __syncthreads();

        // ---- 8 WMMA k-steps of K=4 each ----
        #pragma unroll
        for (int kk = 0; kk < BK; kk += 4) {
            // A fragment: lane lo -> M row, K = kk + 2*hi + {0,1}: contiguous b64 load
            v2f afrag = *reinterpret_cast<const v2f*>(&As[wy * 16 + lo][kk + 2 * hi]);
            #pragma unroll
            for (int t = 0; t < 4; ++t) {
                // B fragment: pair row (kk+2*hi)/2 = kk/2 + hi, contiguous b64 load
                v2f bfrag = *reinterpret_cast<const v2f*>(
                    &BsP[(kk >> 1) + hi][(wx * 64 + t * 16 + lo) * 2]);
                acc[t] = __builtin_amdgcn_wmma_f32_16x16x4_f32(
                    /*neg_a=*/false, afrag, /*neg_b=*/false, bfrag,
                    /*c_mod=*/(short)0, acc[t],
                    /*reuse_a=*/false, /*reuse_b=*/false);
            }
        }
        __syncthreads();
    }

    // ---- epilogue: C/D layout: VGPR r -> row (r + 8*hi), lane lo -> col ----
    #pragma unroll
    for (int t = 0; t < 4; ++t) {
        #pragma unroll
        for (int r = 0; r < 8; ++r) {
            int row = m0 + wy * 16 + r + 8 * hi;
            int col = n0 + wx * 64 + t * 16 + lo;
            float v = acc[t][r];
            if (positions != nullptr) {
                int p = positions[row];
                v += pos_table[(long long)p * N + col];
            }
            C[(long long)row * N + col] = v;
        }
    }
}

// f1[row] = Wh[row,:] . a[0:512]; f2[row] = Wh[row,:] . a[512:1024]
__global__ __launch_bounds__(256)
void f1f2_kernel(const float* __restrict__ Wh, const float* __restrict__ a,
                 float* __restrict__ f1, float* __restrict__ f2)
{
    __shared__ float s1[256], s2[256];
    const int row = blockIdx.x;
    const int t = threadIdx.x;
    const float* wh = Wh + (long long)row * 512;
    float x0 = wh[t], x1 = wh[t + 256];
    s1[t] = x0 * a[t] + x1 * a[t + 256];
    s2[t] = x0 * a[512 + t] + x1 * a[768 + t];
    __syncthreads();
    for (int off = 128; off > 0; off >>= 1) {
        if (t < off) { s1[t] += s1[t + off]; s2[t] += s2[t + off]; }
        __syncthreads();
    }
    if (t == 0) { f1[row] = s1[0]; f2[row] = s2[0]; }
}

// One block per (i, b) attention row: masked leaky-relu logits + softmax.
__global__ __launch_bounds__(256)
void attn_softmax_kernel(const float* __restrict__ f1, const float* __restrict__ f2,
                         const int* __restrict__ adj, float* __restrict__ attn)
{
    const int i = blockIdx.x;    // 0..1023
    const int b = blockIdx.y;    // 0..7
    const int t = threadIdx.x;
    const long long rowBase = ((long long)b * 1024 + i) * 1024;
    const float fi = f1[b * 1024 + i];
    const float* F2 = f2 + b * 1024;
    const int* adjRow = adj + rowBase;

    float e[4];
    #pragma unroll
    for (int q = 0; q < 4; ++q) {
        int j = t + q * 256;
        float x = fi + F2[j];
        x = (x >= 0.0f) ? x : 0.2f * x;          // leaky_relu slope 0.2
        e[q] = (adjRow[j] > 0) ? x : -9.0e15f;   // NEG_INF mask
    }

    __shared__ float sred[256];
    float mx = fmaxf(fmaxf(e[0], e[1]), fmaxf(e[2], e[3]));
    sred[t] = mx; __syncthreads();
    for (int off = 128; off > 0; off >>= 1) {
        if (t < off) sred[t] = fmaxf(sred[t], sred[t + off]);
        __syncthreads();
    }
    mx = sred[0];
    __syncthreads();

    float ex[4], ssum = 0.0f;
    #pragma unroll
    for (int q = 0; q < 4; ++q) { ex[q] = expf(e[q] - mx); ssum += ex[q]; }
    sred[t] = ssum; __syncthreads();
    for (int off = 128; off > 0; off >>= 1) {
        if (t < off) sred[t] += sred[t + off];
        __syncthreads();
    }
    const float inv = 1.0f / sred[0];

    #pragma unroll
    for (int q = 0; q < 4; ++q)
        attn[rowBase + t + q * 256] = ex[q] * inv;
}

extern "C" void kernel_launch(void* const* d_in, const int* in_sizes, int n_in,
                              void* d_out, int out_size, void* d_ws, size_t ws_size,
                              hipStream_t stream) {
    // setup_inputs order: h, adj, positions, W, a, pos_table
    const float* h         = (const float*)d_in[0];  // 8*1024*768
    const int*   adj       = (const int*)  d_in[1];  // 8*1024*1024
    const int*   positions = (const int*)  d_in[2];  // 8*1024
    const float* W         = (const float*)d_in[3];  // 768*512
    const float* a         = (const float*)d_in[4];  // 1024
    const float* pos_table = (const float*)d_in[5];  // 1024*512

    float* h_prime = (float*)d_out;                              // 8*1024*512
    float* attn    = (float*)d_out + (size_t)8 * 1024 * 512;     // 8*1024*1024

    float* Wh = (float*)d_ws;                                    // 8192*512
    float* f1 = Wh + (size_t)8192 * 512;                         // 8192
    float* f2 = f1 + 8192;                                       // 8192

    // 1) Wh = h @ W + pos_table[positions]   (M=8192, N=512, K=768)
    dim3 g1(8192 / BM, 512 / BN, 1);
    gemm_wmma_f32<<<g1, 256, 0, stream>>>(h, W, Wh, 8192, 512, 768,
                                          0LL, 0LL, 0LL, positions, pos_table);

    // 2) f1 = Wh @ a1, f2 = Wh @ a2
    f1f2_kernel<<<dim3(8192, 1, 1), 256, 0, stream>>>(Wh, a, f1, f2);

    // 3) attention = softmax(mask(leaky(f1_i + f2_j)))
    attn_softmax_kernel<<<dim3(1024, 8, 1), 256, 0, stream>>>(f1, f2, adj, attn);

    // 4) h_prime[b] = attention[b] @ Wh[b]   (M=1024, N=512, K=1024, batched over b)
    dim3 g4(1024 / BM, 512 / BN, 8);
    gemm_wmma_f32<<<g4, 256, 0, stream>>>(attn, Wh, h_prime, 1024, 512, 1024,
                                          (long long)1024 * 1024,
                                          (long long)1024 * 512,
                                          (long long)1024 * 512,
                                          nullptr, nullptr);
}